// NeuralNetGlobalHammerWiener_29712583753777
// MI455X (gfx1250) — compile-verified
//
#include <hip/hip_runtime.h>
#include <math.h>

// MI455X / gfx1250, wave32. fp32 end-to-end to match the reference.
typedef __attribute__((ext_vector_type(2))) float v2f;
typedef __attribute__((ext_vector_type(8))) float v8f;

#define W_IN    262144
#define W_OUT   262113        // W_IN - 31
#define TAPS    32
#define CHUNK   2048          // outputs per workgroup
#define HALO    31
#define SLEN    (CHUNK + 32)  // 2080 >= 2079 samples needed (tile K-window = 16*15+48)
#define NCHUNK  128           // ceil(W_OUT / CHUNK)
#define NBATCH  16

__device__ __forceinline__ float fast_rcp(float x) { return __builtin_amdgcn_rcpf(x); }

__device__ __forceinline__ float fast_tanh(float x) {
  // tanh(x) = 1 - 2/(exp(2x)+1); v_exp_f32 + v_rcp_f32 on the TRANS pipe
  float e = __expf(2.0f * x);
  return 1.0f - 2.0f * fast_rcp(e + 1.0f);
}

__global__ __launch_bounds__(256)
void hammer_wiener_wmma(const float* __restrict__ xr_g, const float* __restrict__ xi_g,
                        const float* __restrict__ w1_pre, const float* __restrict__ w2_pre,
                        const float* __restrict__ wfr,    const float* __restrict__ wfi,
                        const float* __restrict__ w1_post,const float* __restrict__ b1_post,
                        const float* __restrict__ w2_post,const float* __restrict__ b2_post,
                        float* __restrict__ out)
{
  __shared__ float s0[SLEN];     // xh_r
  __shared__ float s1[SLEN];     // xh_i
  __shared__ float zr_s[CHUNK];  // FIR real out
  __shared__ float zi_s[CHUNK];  // FIR imag out

  const int tid = threadIdx.x;
  const int b   = blockIdx.y;
  const int c0  = blockIdx.x * CHUNK;
  const float* xr = xr_g + (size_t)b * W_IN;
  const float* xi = xi_g + (size_t)b * W_IN;

  // ---- stage 1: pre-MLP (1->8 tanh ->1) + phase rotation, into LDS ----
  {
    float w1p[8], w2p[8];
#pragma unroll
    for (int j = 0; j < 8; ++j) { w1p[j] = w1_pre[j]; w2p[j] = w2_pre[j]; }

    for (int i = tid; i < CHUNK + HALO; i += 256) {
      int win = c0 + i; if (win > W_IN - 1) win = W_IN - 1;  // halo clamp (only discarded outputs touched)
      float r = xr[win], im = xi[win];
      float mag = sqrtf(r * r + im * im);
      float acc = 0.0f;
#pragma unroll
      for (int j = 0; j < 8; ++j) acc += fast_tanh(mag * w1p[j]) * w2p[j];
      bool nz = mag > 0.0f;
      float inv = nz ? fast_rcp(mag) : 0.0f;
      float cp = nz ? r * inv : 1.0f;   // cos(atan2(0,0)) = 1
      float sp = nz ? im * inv : 0.0f;
      s0[i] = acc * cp;
      s1[i] = acc * sp;
    }
  }
  __syncthreads();

  // ---- stage 2: complex 32-tap FIR via V_WMMA_F32_16X16X4_F32 ----
  // A = 16x48 Toeplitz of filter taps: A[m,k] = w[k-m], K split into 12 chunks of 4.
  // B[k,n] = s[tb + 16n + k]  ->  D[m,n] = y[tb + 16n + m] (256 contiguous outputs/tile).
  {
    const int lane = tid & 31;
    const int wave = tid >> 5;
    const int mrow = lane & 15;          // A row m / B-D column n
    const int kh   = lane >> 4;          // lane half selects K pair {0,1} vs {2,3}
    const int kx   = kh << 1;

    v2f awr[12], awi[12];
#pragma unroll
    for (int kc = 0; kc < 12; ++kc) {
      int k0 = 4 * kc + kx;
      int t0 = k0 - mrow, t1 = k0 + 1 - mrow;
      awr[kc].x = (t0 >= 0 && t0 < TAPS) ? wfr[t0] : 0.0f;
      awr[kc].y = (t1 >= 0 && t1 < TAPS) ? wfr[t1] : 0.0f;
      awi[kc].x = (t0 >= 0 && t0 < TAPS) ? wfi[t0] : 0.0f;
      awi[kc].y = (t1 >= 0 && t1 < TAPS) ? wfi[t1] : 0.0f;
    }

    const int tb  = wave * 256;          // each wave owns one 256-output tile
    const int col = tb + 16 * mrow;      // base of this lane's B column
    v8f accR = {};                       // zr = wr*s0 - wi*s1
    v8f accI = {};                       // zi = wi*s0 + wr*s1
#pragma unroll
    for (int kc = 0; kc < 12; ++kc) {
      int kB = 4 * kc + kx;
      v2f b0, b1, nwi;
      b0.x = s0[col + kB]; b0.y = s0[col + kB + 1];
      b1.x = s1[col + kB]; b1.y = s1[col + kB + 1];
      nwi.x = -awi[kc].x;  nwi.y = -awi[kc].y;
      accR = __builtin_amdgcn_wmma_f32_16x16x4_f32(false, awr[kc], false, b0, (short)0, accR, false, false);
      accR = __builtin_amdgcn_wmma_f32_16x16x4_f32(false, nwi,     false, b1, (short)0, accR, false, false);
      accI = __builtin_amdgcn_wmma_f32_16x16x4_f32(false, awi[kc], false, b0, (short)0, accI, false, false);
      accI = __builtin_amdgcn_wmma_f32_16x16x4_f32(false, awr[kc], false, b1, (short)0, accI, false, false);
    }

    // D layout: VGPR r holds M=r (lanes 0-15) / M=8+r (lanes 16-31), N = lane&15
    const int mb = kh * 8;
#pragma unroll
    for (int r = 0; r < 8; ++r) {
      int o = col + mb + r;              // tb + 16n + m
      zr_s[o] = accR[r];
      zi_s[o] = accI[r];
    }
  }
  __syncthreads();

  // ---- stage 3: post-MLP (1->8 relu ->1), rescale by phase, store ----
  {
    float w1q[8], b1q[8], w2q[8];
#pragma unroll
    for (int j = 0; j < 8; ++j) { w1q[j] = w1_post[j]; b1q[j] = b1_post[j]; w2q[j] = w2_post[j]; }
    const float b2 = b2_post[0];
    const float SCALE = 0.17782794100389228f;  // sqrt(10^-1.5)

    for (int o = tid; o < CHUNK; o += 256) {
      int wout = c0 + o;
      if (wout < W_OUT) {
        float zr = zr_s[o], zi = zi_s[o];
        float zm = sqrtf(zr * zr + zi * zi);
        float g = b2;
#pragma unroll
        for (int j = 0; j < 8; ++j) {
          float h = zm * w1q[j] + b1q[j];
          g += (h > 0.0f ? h : 0.0f) * w2q[j];
        }
        bool nz = zm > 0.0f;
        float inv = nz ? fast_rcp(zm) : 0.0f;
        float cp = nz ? zr * inv : 1.0f;
        float sp = nz ? zi * inv : 0.0f;
        size_t oidx = ((size_t)b * W_OUT + (size_t)wout) * 2;
        out[oidx]     = SCALE * g * cp;
        out[oidx + 1] = SCALE * g * sp;
      }
    }
  }
}

extern "C" void kernel_launch(void* const* d_in, const int* in_sizes, int n_in,
                              void* d_out, int out_size, void* d_ws, size_t ws_size,
                              hipStream_t stream) {
  (void)in_sizes; (void)n_in; (void)out_size; (void)d_ws; (void)ws_size;
  const float* xr      = (const float*)d_in[0];
  const float* xi      = (const float*)d_in[1];
  const float* w1_pre  = (const float*)d_in[2];
  const float* w2_pre  = (const float*)d_in[3];
  const float* w_fir_r = (const float*)d_in[4];
  const float* w_fir_i = (const float*)d_in[5];
  const float* w1_post = (const float*)d_in[6];
  const float* b1_post = (const float*)d_in[7];
  const float* w2_post = (const float*)d_in[8];
  const float* b2_post = (const float*)d_in[9];
  float* out = (float*)d_out;

  dim3 grid(NCHUNK, NBATCH);
  hammer_wiener_wmma<<<grid, 256, 0, stream>>>(xr, xi, w1_pre, w2_pre,
                                               w_fir_r, w_fir_i,
                                               w1_post, b1_post, w2_post, b2_post,
                                               out);
}